// CurvConv_68212670595579
// MI455X (gfx1250) — compile-verified
//
#include <hip/hip_runtime.h>

typedef float v2f __attribute__((ext_vector_type(2)));
typedef float v8f __attribute__((ext_vector_type(8)));

#define N_NODES 10000
#define N_EDGES 640000
#define DIM     128

// ---------------------------------------------------------------------------
// Kernel 1: zero the output accumulator (float4 vectorized)
// ---------------------------------------------------------------------------
__global__ void cc_zero_f4(float4* __restrict__ p, int n4) {
    int i = blockIdx.x * blockDim.x + threadIdx.x;
    if (i < n4) p[i] = make_float4(0.f, 0.f, 0.f, 0.f);
}

// ---------------------------------------------------------------------------
// Kernel 2: per-edge curvature weight
//   c[e] = lin_b + sum_{k=1..10} lin_w[k-1] * (1 + exp(-curva[e]*k)) * 0.5
// ---------------------------------------------------------------------------
__global__ void cc_edge_curv(const float* __restrict__ curva,
                             const float* __restrict__ lin_w,
                             const float* __restrict__ lin_b,
                             float* __restrict__ cw) {
    int e = blockIdx.x * blockDim.x + threadIdx.x;
    if (e >= N_EDGES) return;
    float t = curva[e];
    float acc = lin_b[0];
#pragma unroll
    for (int k = 1; k <= 10; ++k) {
        acc += lin_w[k - 1] * (1.0f + expf(-t * (float)k)) * 0.5f;
    }
    cw[e] = acc;
}

// ---------------------------------------------------------------------------
// Kernel 3: h = x @ W + b via V_WMMA_F32_16X16X4_F32 (fp32-exact)
//   grid.x = 625 (16-row strips of x), blockDim = (32, 8): 8 waves, one per
//   16-column tile of W. Each wave runs a K-loop of 32 WMMA ops (K=128, 4/step).
//
// VGPR layouts per CDNA5 ISA 7.12.2 (32-bit, 16x4 A / 4x16 B / 16x16 C):
//   A: lane<16 -> M=lane, {v0,v1}={K+0,K+1}; lane>=16 -> M=lane-16, {K+2,K+3}
//   B: lane<16 -> N=lane, {v0,v1}={K+0,K+1} rows; lane>=16 -> rows {K+2,K+3}
//   C/D: VGPR v -> M = v + 8*(lane>=16), N = lane&15
// ---------------------------------------------------------------------------
__global__ void cc_gemm_wmma(const float* __restrict__ x,
                             const float* __restrict__ W,
                             const float* __restrict__ bias,
                             float* __restrict__ h) {
    const int lane  = threadIdx.x;        // 0..31
    const int r0    = blockIdx.x * 16;    // row strip of x / h
    const int c0    = threadIdx.y * 16;   // column tile of W / h
    const int idx16 = lane & 15;
    const int half  = lane >> 4;          // 0 or 1
    const int koff  = half * 2;           // K sub-offset for this lane half

    v8f acc = {};

    const float* xrow = x + (r0 + idx16) * DIM;   // A-matrix row M = lane&15
    const float* wcol = W + c0 + idx16;           // B-matrix col N = lane&15

#pragma unroll
    for (int k = 0; k < DIM; k += 4) {
        v2f a;
        a.x = xrow[k + koff];
        a.y = xrow[k + koff + 1];
        v2f b;
        b.x = wcol[(k + koff) * DIM];
        b.y = wcol[(k + koff + 1) * DIM];
        // 8 args: (neg_a, A, neg_b, B, c_mod, C, reuse_a, reuse_b)
        acc = __builtin_amdgcn_wmma_f32_16x16x4_f32(
            false, a, false, b, (short)0, acc, false, false);
    }

    const float bv = bias[c0 + idx16];
#pragma unroll
    for (int v = 0; v < 8; ++v) {
        const int m = v + half * 8;
        h[(r0 + m) * DIM + c0 + idx16] = acc[v] + bv;
    }
}

// ---------------------------------------------------------------------------
// Kernel 4: edge message passing. One wave per edge.
//   Each lane gathers 4 consecutive floats of h[src] (global_load_b128),
//   scales by the edge curvature weight, and does 4 coalesced
//   global_atomic_add_f32 into out[dst]. out (5.1 MB) and h (5.1 MB) are
//   fully L2-resident -> this phase is L2-atomic-throughput bound.
// ---------------------------------------------------------------------------
__global__ void cc_edge_scatter(const float* __restrict__ h,
                                const float* __restrict__ cw,
                                const int* __restrict__ src,
                                const int* __restrict__ dst,
                                float* __restrict__ out) {
    const int lane = threadIdx.x & 31;
    const int e = blockIdx.x * (blockDim.x >> 5) + (threadIdx.x >> 5);
    if (e >= N_EDGES) return;

    const int   s = src[e];
    const int   d = dst[e];
    const float c = cw[e];

    const float4 v = *reinterpret_cast<const float4*>(h + s * DIM + lane * 4);
    float* o = out + d * DIM + lane * 4;
    atomicAdd(o + 0, v.x * c);
    atomicAdd(o + 1, v.y * c);
    atomicAdd(o + 2, v.z * c);
    atomicAdd(o + 3, v.w * c);
}

// ---------------------------------------------------------------------------
// Kernel 5: in-place ReLU on the accumulated output (float4 vectorized)
// ---------------------------------------------------------------------------
__global__ void cc_relu_f4(float4* __restrict__ p, int n4) {
    int i = blockIdx.x * blockDim.x + threadIdx.x;
    if (i < n4) {
        float4 v = p[i];
        v.x = fmaxf(v.x, 0.f);
        v.y = fmaxf(v.y, 0.f);
        v.z = fmaxf(v.z, 0.f);
        v.w = fmaxf(v.w, 0.f);
        p[i] = v;
    }
}

// ---------------------------------------------------------------------------
// Host-side launch. Inputs (setup_inputs order):
//   0: x        [N_NODES*DIM]  f32
//   1: curva    [N_EDGES]      f32
//   2: edge_idx [2*N_EDGES]    int
//   3: W        [DIM*DIM]      f32
//   4: b        [DIM]          f32
//   5: lin_w    [10]           f32
//   6: lin_b    [1]            f32
// Output: [N_NODES*DIM] f32
// Workspace: h (N_NODES*DIM f32) | cw (N_EDGES f32)  = 7.68 MB
// ---------------------------------------------------------------------------
extern "C" void kernel_launch(void* const* d_in, const int* in_sizes, int n_in,
                              void* d_out, int out_size, void* d_ws, size_t ws_size,
                              hipStream_t stream) {
    const float* x     = (const float*)d_in[0];
    const float* curva = (const float*)d_in[1];
    const int*   eidx  = (const int*)d_in[2];
    const float* W     = (const float*)d_in[3];
    const float* b     = (const float*)d_in[4];
    const float* lin_w = (const float*)d_in[5];
    const float* lin_b = (const float*)d_in[6];
    float*       out   = (float*)d_out;

    float* h  = (float*)d_ws;              // N_NODES*DIM floats
    float* cw = h + (size_t)N_NODES * DIM; // N_EDGES floats

    const int* src = eidx;                 // edge_index[0, :]
    const int* dst = eidx + N_EDGES;       // edge_index[1, :]

    // 1) zero accumulator (d_out)
    {
        int n4 = (N_NODES * DIM) / 4;                       // 320000
        cc_zero_f4<<<(n4 + 255) / 256, 256, 0, stream>>>((float4*)out, n4);
    }
    // 2) per-edge curvature weights
    cc_edge_curv<<<(N_EDGES + 255) / 256, 256, 0, stream>>>(curva, lin_w, lin_b, cw);
    // 3) WMMA GEMM: h = x @ W + b
    {
        dim3 blk(32, 8);                                    // 8 waves = 8 col tiles
        cc_gemm_wmma<<<N_NODES / 16, blk, 0, stream>>>(x, W, b, h);
    }
    // 4) gather/scale/scatter-add over edges (wave per edge)
    {
        int waves_per_block = 256 / 32;                     // 8 edges per block
        int blocks = (N_EDGES + waves_per_block - 1) / waves_per_block;
        cc_edge_scatter<<<blocks, 256, 0, stream>>>(h, cw, src, dst, out);
    }
    // 5) ReLU in place
    {
        int n4 = (N_NODES * DIM) / 4;
        cc_relu_f4<<<(n4 + 255) / 256, 256, 0, stream>>>((float4*)out, n4);
    }
}